// CenterDirGroundtruth_67602785239349
// MI455X (gfx1250) — compile-verified
//
#include <hip/hip_runtime.h>
#include <hip/hip_bf16.h>
#include <cmath>

#define MAXC 100          // MAX_NUM_CENTERS
#define KS   21           // gaussian kernel size
#define KR   10           // radius
#define NCH  13

typedef float v2f __attribute__((ext_vector_type(2)));
typedef float v8f __attribute__((ext_vector_type(8)));

// ---------------------------------------------------------------- utilities

__device__ __forceinline__ int refl(int i, int n) {
    // jnp.pad 'reflect' (no edge repeat): -1 -> 1, n -> n-2
    if (i < 0)  i = -i;
    if (i >= n) i = 2 * n - 2 - i;
    return i;
}

// cooperatively build dense banded gaussian CB[t*16+n] = g[t-n] (36x16)
// serves as B-matrix of the row pass AND A-matrix of the col pass.
__device__ __forceinline__ void build_band(float* CB) {
    float sum = 0.0f;
    for (int j = 0; j < KS; ++j) {
        float h = (float)(j - KR) * 0.5f;            // x / SIGMA, SIGMA=2
        sum += expf(-0.5f * h * h);
    }
    for (int idx = threadIdx.x; idx < 36 * 16; idx += blockDim.x) {
        int t = idx >> 4, n = idx & 15;
        int d = t - n;
        float h = (float)(d - KR) * 0.5f;
        CB[idx] = (d >= 0 && d < KS) ? expf(-0.5f * h * h) / sum : 0.0f;
    }
}

// ---------------------------------------------------------------- kernels

__global__ void init_ws_kernel(int* ws, int n) {
    int i = blockIdx.x * blockDim.x + threadIdx.x;
    if (i < n) ws[i] = 0;
}

// presence bitmap per batch + has_inst flag
__global__ void presence_kernel(const int* __restrict__ inst,
                                int* __restrict__ present,
                                int* __restrict__ has_inst, int HW) {
    __shared__ int pl[MAXC];
    int b = blockIdx.y;
    for (int k = threadIdx.x; k < MAXC; k += blockDim.x) pl[k] = 0;
    __syncthreads();
    const int* ip = inst + (size_t)b * HW;
    for (int i = blockIdx.x * blockDim.x + threadIdx.x; i < HW;
         i += gridDim.x * blockDim.x) {
        int v = ip[i];
        v = v < 0 ? 0 : (v >= MAXC ? MAXC - 1 : v);
        pl[v] = 1;                                   // benign race: all write 1
    }
    __syncthreads();
    for (int k = threadIdx.x; k < MAXC; k += blockDim.x)
        if (pl[k]) atomicOr(&present[b * MAXC + k], 1);
    if (threadIdx.x == 0) {
        int any = 0;
        for (int k = 1; k < MAXC; ++k) any |= pl[k];
        if (any) atomicOr(&has_inst[b], 1);
    }
}

// per-pixel math, 4 px/thread, float4 stores (store-stream dominated kernel)
__global__ void pixel_kernel(const int* __restrict__ inst,
                             const float* __restrict__ centers,
                             const int* __restrict__ has_inst,
                             int* __restrict__ any_near,
                             float* __restrict__ mat, int H, int W) {
    int b = blockIdx.y;
    int q = blockIdx.x * blockDim.x + threadIdx.x;   // quad index
    int HW = H * W;
    if (q >= (HW >> 2)) return;
    int i = q << 2;
    int y = i / W, x0 = i - y * W;                   // W % 4 == 0: same row
    bool hi = has_inst[b] != 0;

    const int4 vi = *reinterpret_cast<const int4*>(inst + (size_t)b * HW + i);
    const float* cb = centers + b * (MAXC * 2);      // per-batch centers base
    int vv[4] = {vi.x, vi.y, vi.z, vi.w};
    float rR[4], rT[4], rS[4], rC[4], rG[4], rN[4];
    int anyn = 0;
#pragma unroll
    for (int k = 0; k < 4; ++k) {
        int v = vv[k];
        bool valid = v > 0;
        int idx = valid ? v - 1 : 0;
        float cy = cb[idx * 2 + 0];
        float cx = cb[idx * 2 + 1];
        float gx = valid ? cx : -10000.0f;           // gt_cx
        float gy = valid ? cy : -10000.0f;           // gt_cy
        float X = gx - (float)y;                     // gt_cx - row
        float Y = gy - (float)(x0 + k);              // gt_cy - col
        bool nearC = (fabsf(X) < 3.0f) && (fabsf(Y) < 3.0f);
        float ignore = hi ? (nearC ? 0.0f : 1.0f) : 1.0f;
        float m = valid ? 1.0f : 0.0f;
        float th = atan2f(Y, X);
        float s = sinf(th) * m;
        float c = cosf(th) * m;
        float M = sqrtf(s * s + c * c);
        float invM = 1.0f / fmaxf(M, 1e-12f);
        rR[k] = sqrtf(X * X + Y * Y) * m;
        rT[k] = hi ? th : 0.0f;
        rS[k] = valid ? s * invM : 0.0f;
        rC[k] = valid ? c * invM : 0.0f;
        rG[k] = ignore;
        rN[k] = 1.0f - ignore;                       // blur input
        anyn |= (int)(nearC && hi);
    }
    if (anyn) atomicOr(&any_near[b], 1);             // ~25 px/img -> cheap

    float* base = mat + (size_t)b * NCH * HW;        // 64-bit once
    float4* p;
    p = (float4*)(base + 0 * HW + i); *p = make_float4(rR[0], rR[1], rR[2], rR[3]);
    p = (float4*)(base + 1 * HW + i); *p = make_float4(rT[0], rT[1], rT[2], rT[3]);
    p = (float4*)(base + 2 * HW + i); *p = make_float4(rS[0], rS[1], rS[2], rS[3]);
    p = (float4*)(base + 3 * HW + i); *p = make_float4(rC[0], rC[1], rC[2], rC[3]);
    p = (float4*)(base + 4 * HW + i); *p = make_float4(rG[0], rG[1], rG[2], rG[3]);
    p = (float4*)(base + 5 * HW + i); *p = make_float4(rN[0], rN[1], rN[2], rN[3]);
    float4 z = make_float4(0.f, 0.f, 0.f, 0.f);
#pragma unroll
    for (int ch = 7; ch < NCH; ++ch) {
        p = (float4*)(base + ch * HW + i); *p = z;
    }
}

// horizontal blur: Out(16x16) = In(16x36) x CB(36x16) via 9 x WMMA f32 16x16x4
// reads ch5, writes ch6
__global__ void blur_row_kernel(float* __restrict__ mat, int H, int W) {
    __shared__ float CB[36 * 16];
    __shared__ float tile[4][16 * 36];
    build_band(CB);
    int wave = threadIdx.x >> 5;
    int lane = threadIdx.x & 31;
    int half = lane >> 4, l16 = lane & 15;
    int b  = blockIdx.z;
    int y0 = blockIdx.y * 16;
    int x0 = (blockIdx.x * 4 + wave) * 16;
    int HW = H * W;
    const float* src = mat + (size_t)(b * NCH + 5) * HW;   // 64-bit once
    float*       dst = mat + (size_t)(b * NCH + 6) * HW;
    float* T = tile[wave];
    for (int idx = lane; idx < 16 * 36; idx += 32) { // full wave: no divergence
        int r = idx / 36, c = idx - r * 36;
        T[idx] = src[(y0 + r) * W + refl(x0 - KR + c, W)]; // 32-bit offsets
    }
    __syncthreads();
    v8f acc = {};
#pragma unroll
    for (int ci = 0; ci < 9; ++ci) {
        int t0 = 4 * ci + 2 * half;
        v2f a, bb;
        a.x  = T[l16 * 36 + t0];                     // A: data (contiguous pair)
        a.y  = T[l16 * 36 + t0 + 1];
        bb.x = CB[t0 * 16 + l16];                    // B: banded gaussian
        bb.y = CB[(t0 + 1) * 16 + l16];
        acc = __builtin_amdgcn_wmma_f32_16x16x4_f32(
            false, a, false, bb, (short)0, acc, false, false);
    }
    int o = (y0 + 8 * half) * W + x0 + l16;          // 32-bit store offsets
#pragma unroll
    for (int v = 0; v < 8; ++v)
        dst[o + v * W] = acc[v];
}

// vertical blur: Out(16x16) = CB-band(16x36) x In(36x16); + per-batch max
// reads ch6, writes ch5
__global__ void blur_col_kernel(float* __restrict__ mat,
                                unsigned int* __restrict__ bmaxU,
                                int H, int W) {
    __shared__ float CB[36 * 16];
    __shared__ float tile[4][36 * 16];
    build_band(CB);
    int wave = threadIdx.x >> 5;
    int lane = threadIdx.x & 31;
    int half = lane >> 4, l16 = lane & 15;
    int b  = blockIdx.z;
    int y0 = blockIdx.y * 16;
    int x0 = (blockIdx.x * 4 + wave) * 16;
    int HW = H * W;
    const float* src = mat + (size_t)(b * NCH + 6) * HW;
    float*       dst = mat + (size_t)(b * NCH + 5) * HW;
    float* T = tile[wave];
    for (int idx = lane; idx < 36 * 16; idx += 32) {
        int r = idx >> 4, c = idx & 15;
        T[idx] = src[refl(y0 - KR + r, H) * W + x0 + c];   // 32-bit offsets
    }
    __syncthreads();
    v8f acc = {};
#pragma unroll
    for (int ci = 0; ci < 9; ++ci) {
        int t0 = 4 * ci + 2 * half;
        v2f a, bb;
        a.x  = CB[t0 * 16 + l16];                    // A: banded gaussian
        a.y  = CB[(t0 + 1) * 16 + l16];
        bb.x = T[t0 * 16 + l16];                     // B: data
        bb.y = T[(t0 + 1) * 16 + l16];
        acc = __builtin_amdgcn_wmma_f32_16x16x4_f32(
            false, a, false, bb, (short)0, acc, false, false);
    }
    float mx = 0.0f;
    int o = (y0 + 8 * half) * W + x0 + l16;          // 32-bit store offsets
#pragma unroll
    for (int v = 0; v < 8; ++v) {
        dst[o + v * W] = acc[v];
        mx = fmaxf(mx, acc[v]);
    }
    for (int off = 16; off > 0; off >>= 1)
        mx = fmaxf(mx, __shfl_xor(mx, off, 32));
    if (lane == 0)
        atomicMax(&bmaxU[b], __float_as_uint(mx));   // blur >= 0: uint order ok
}

// cmask = all_ones ? 0 : blur / max(bmax, 1e-12); zero ch6 (float4)
__global__ void finalize_kernel(float* __restrict__ mat,
                                const int* __restrict__ has_inst,
                                const int* __restrict__ any_near,
                                const unsigned int* __restrict__ bmaxU, int HW) {
    int b = blockIdx.y;
    int q = blockIdx.x * blockDim.x + threadIdx.x;
    if (q >= (HW >> 2)) return;
    int i = q << 2;
    bool all_ones = !(has_inst[b] && any_near[b]);
    float bm = __uint_as_float(bmaxU[b]);
    float den = fmaxf(bm, 1e-12f);
    float* base = mat + (size_t)b * NCH * HW;        // 64-bit once
    float4* p5 = (float4*)(base + 5 * HW + i);
    float4* p6 = (float4*)(base + 6 * HW + i);
    float4 bl = *p5;
    float4 cm;
    cm.x = all_ones ? 0.0f : bl.x / den;
    cm.y = all_ones ? 0.0f : bl.y / den;
    cm.z = all_ones ? 0.0f : bl.z / den;
    cm.w = all_ones ? 0.0f : bl.w / den;
    *p5 = cm;
    *p6 = make_float4(0.f, 0.f, 0.f, 0.f);
}

__global__ void centers_kernel(const float* __restrict__ centers,
                               const int* __restrict__ present,
                               float* __restrict__ out, int B) {
    int i = blockIdx.x * blockDim.x + threadIdx.x;
    if (i >= B * MAXC) return;
    int b = i / MAXC, j = i - b * MAXC;
    float o0 = 0.0f, o1 = 0.0f;
    if (j >= 1 && present[b * MAXC + j]) {
        o0 = centers[(b * MAXC + (j - 1)) * 2 + 1];  // swapped
        o1 = centers[(b * MAXC + (j - 1)) * 2 + 0];
    }
    out[i * 2 + 0] = o0;
    out[i * 2 + 1] = o1;
}

// ---------------------------------------------------------------- launch

extern "C" void kernel_launch(void* const* d_in, const int* in_sizes, int n_in,
                              void* d_out, int out_size, void* d_ws, size_t ws_size,
                              hipStream_t stream) {
    const int*   inst    = (const int*)d_in[0];
    const float* centers = (const float*)d_in[1];
    float*       out     = (float*)d_out;

    int B  = in_sizes[1] / (2 * MAXC);               // centers: (B,100,2)
    int HW = in_sizes[0] / B;                        // instances: (B,1,H,W)
    int H  = (int)(sqrt((double)HW) + 0.5);          // square per reference
    int W  = H;

    // workspace layout (ints): present[B*100] | has_inst[B] | any_near[B] | bmax[B]
    int* ws        = (int*)d_ws;
    int* present   = ws;
    int* has_inst  = ws + B * MAXC;
    int* any_near  = has_inst + B;
    unsigned int* bmaxU = (unsigned int*)(any_near + B);
    int nInit = B * MAXC + 3 * B;

    init_ws_kernel<<<(nInit + 255) / 256, 256, 0, stream>>>(ws, nInit);
    presence_kernel<<<dim3(64, B), 256, 0, stream>>>(inst, present, has_inst, HW);

    int nq = HW >> 2;                                // 4 px per thread
    pixel_kernel<<<dim3((nq + 255) / 256, B), 256, 0, stream>>>(
        inst, centers, has_inst, any_near, out, H, W);

    dim3 bg(W / 64, H / 16, B);                      // 4 waves/block, 1 tile/wave
    blur_row_kernel<<<bg, 128, 0, stream>>>(out, H, W);
    blur_col_kernel<<<bg, 128, 0, stream>>>(out, bmaxU, H, W);

    finalize_kernel<<<dim3((nq + 255) / 256, B), 256, 0, stream>>>(
        out, has_inst, any_near, bmaxU, HW);

    float* gt = out + (size_t)B * NCH * HW;
    centers_kernel<<<(B * MAXC + 255) / 256, 256, 0, stream>>>(centers, present, gt, B);
}